// SparselyGatedDotLS_56504589746311
// MI455X (gfx1250) — compile-verified
//
#include <hip/hip_runtime.h>
#include <math.h>

// ---------------------------------------------------------------------------
// SparselyGatedDotLS for MI455X (gfx1250, wave32).
// Memory-bound problem (~540 MB traffic @ 23.3 TB/s => ~23us floor).
// Gate reduction runs through V_WMMA_F32_16X16X4_F32:
//   A (16x4)  = gate-weighted nmt "w" chunk, broadcast to all 16 M-rows
//   B (4x16)  = pooled hidden-state values, column c = expert/state index
//   D row 0   = 16-wide gate partial vector (experts in the N dimension)
// ---------------------------------------------------------------------------

typedef __attribute__((ext_vector_type(2))) float v2f;
typedef __attribute__((ext_vector_type(8))) float v8f;

#define N_STATES 13
#define T_DIM    1024
#define B_DIM    8
#define D_DIM    1024
#define DN_DIM   512
#define ROWP     516          // LDS row pitch (floats); 516 % 64 = 4 -> conflict-free b64 loads
#define TCHUNK   8
#define NCHUNKS  (T_DIM / TCHUNK)   // 128

// ---------------- Phase A: gate partial sums (reads all of hidden_states) --
__global__ __launch_bounds__(128) void gate_partial_kernel(
    const float* __restrict__ hs,    // [13,1024,8,1024]
    const float* __restrict__ nmt,   // [2048,8,512]
    const float* __restrict__ gw,    // [512]
    float* __restrict__ partial)     // [B_DIM*NCHUNKS, 16]
{
  __shared__ __align__(16) float wlds[DN_DIM];
  __shared__ __align__(16) float hp[16 * ROWP];
  __shared__ float wsum[4][16];

  const int tid  = threadIdx.x;        // 0..127 (4 waves)
  const int lane = tid & 31;
  const int r    = lane & 15;          // N index for B-frag / M index for A
  const int koff = (lane >> 4) << 1;   // K sub-offset: lanes 0-15 -> K0,K1; 16-31 -> K2,K3

  // Wave id as an SGPR: keeps the WMMA loop scalar-controlled, so EXEC is
  // never masked around v_wmma (ISA requires EXEC == all ones for WMMA).
  const int wv = __builtin_amdgcn_readfirstlane(tid >> 5);

  const int bid   = blockIdx.x;        // b*NCHUNKS + chunk
  const int b     = bid >> 7;
  const int chunk = bid & (NCHUNKS - 1);

  // Zero pad expert rows 13..15 once (keeps unused D columns exactly 0).
  for (int i = tid; i < 3 * DN_DIM; i += 128)
    hp[(N_STATES + i / DN_DIM) * ROWP + (i % DN_DIM)] = 0.0f;

  v8f acc = {};   // persistent accumulator: sums over t and this wave's d-range

  for (int tt = 0; tt < TCHUNK; ++tt) {
    const int t = chunk * TCHUNK + tt;
    __syncthreads();  // previous iteration's WMMA readers done before overwrite

    // w[d] = max(nmt[2t,b,d], nmt[2t+1,b,d]) * gw[d]
    {
      const float* r0 = nmt + ((size_t)(2 * t) * B_DIM + b) * DN_DIM;
      const float* r1 = r0 + (size_t)B_DIM * DN_DIM;
      const int d = tid * 4;
      float4 a0 = *(const float4*)(r0 + d);
      float4 a1 = *(const float4*)(r1 + d);
      float4 g  = *(const float4*)(gw + d);
      wlds[d + 0] = fmaxf(a0.x, a1.x) * g.x;
      wlds[d + 1] = fmaxf(a0.y, a1.y) * g.y;
      wlds[d + 2] = fmaxf(a0.z, a1.z) * g.z;
      wlds[d + 3] = fmaxf(a0.w, a1.w) * g.w;
    }

    // Pool hidden_states rows: hp[n][d'] = h[2d'] + h[2d'+1].
    // (The 0.5 mean factor is folded into phase B's final scale.)
    // Fully coalesced float4 streams along D (the 435 MB hot path).
    {
      const float* base = hs + ((size_t)t * B_DIM + b) * D_DIM;
      const int d = tid * 8;
      for (int n = 0; n < N_STATES; ++n) {
        const float* row = base + (size_t)n * T_DIM * B_DIM * D_DIM;
        float4 h0 = *(const float4*)(row + d);
        float4 h1 = *(const float4*)(row + d + 4);
        float* o = hp + n * ROWP + tid * 4;
        o[0] = h0.x + h0.y;
        o[1] = h0.z + h0.w;
        o[2] = h1.x + h1.y;
        o[3] = h1.z + h1.w;
      }
    }
    __syncthreads();

    // WMMA accumulate: each wave covers a 128-wide d-quarter, K=4 per step.
    // afrag: same w pair for all lanes (broadcast LDS read) -> every D row
    //        equals the gate vector; we keep row 0 (lanes 0-15, acc[0]).
    // bfrag: hp[r][d0+koff .. +1] -- one ds_load_b64 per lane, bank-conflict
    //        free thanks to the 516-float row pitch.
    const int dbeg = wv * 128;           // SGPR -> scalar loop bounds
    #pragma unroll 8
    for (int d0 = dbeg; d0 < dbeg + 128; d0 += 4) {
      v2f bfrag = *(const v2f*)(hp + r * ROWP + d0 + koff);
      v2f afrag = *(const v2f*)(wlds + d0 + koff);
      acc = __builtin_amdgcn_wmma_f32_16x16x4_f32(
          false, afrag, false, bfrag, (short)0, acc, false, false);
    }
  }

  // Combine the 4 wave partials (D row 0 lives in acc[0] of lanes 0..15).
  if (lane < 16) wsum[wv][lane] = acc[0];
  __syncthreads();
  if (tid < 16)
    partial[(size_t)bid * 16 + tid] =
        (wsum[0][tid] + wsum[1][tid]) + (wsum[2][tid] + wsum[3][tid]);
}

// ---------------- Phase B: reduce chunks, top-2 + softmax (tiny) -----------
__global__ __launch_bounds__(128) void gate_finish_kernel(
    const float* __restrict__ partial,   // [B_DIM*NCHUNKS, 16]
    const float* __restrict__ gate_b,    // [1]
    float* __restrict__ meta)            // [B_DIM][4]: s0, s1, (int)i0, (int)i1
{
  __shared__ float g[B_DIM][16];
  const int tid = threadIdx.x;      // 128 = 8 b * 16 n
  const int b = tid >> 4;
  const int n = tid & 15;

  float s = 0.0f;                   // fixed-order, deterministic
  for (int c = 0; c < NCHUNKS; ++c)
    s += partial[((size_t)b * NCHUNKS + c) * 16 + n];
  // 0.5 = pooling mean (mean over s=2, deferred from phase A); 1/T = mean over t
  g[b][n] = s * (0.5f / (float)T_DIM) + gate_b[0];
  __syncthreads();

  if (n == 0) {
    int i0 = 0; float v0 = g[b][0];
    for (int k = 1; k < N_STATES; ++k) {
      float v = g[b][k];
      if (v > v0) { v0 = v; i0 = k; }           // strict > : lowest index on ties (jax)
    }
    int i1 = (i0 == 0) ? 1 : 0; float v1 = g[b][i1];
    for (int k = 0; k < N_STATES; ++k) {
      if (k == i0) continue;
      float v = g[b][k];
      if (v > v1) { v1 = v; i1 = k; }
    }
    // softmax over {v0, v1} with v0 >= v1, folding in the mean-over-2 (0.5)
    float e1  = expf(v1 - v0);
    float inv = 0.5f / (1.0f + e1);
    float* mb = meta + b * 4;
    mb[0] = inv;
    mb[1] = e1 * inv;
    ((int*)mb)[2] = i0;
    ((int*)mb)[3] = i1;
  }
}

// ---------------- Phase C: weighted combine of the two selected experts ----
__global__ __launch_bounds__(256) void combine_kernel(
    const float* __restrict__ hs,    // [13,1024,8,1024]
    const float* __restrict__ meta,  // [B_DIM][4]
    float* __restrict__ out)         // [1024,8,1024] (t,b,d)
{
  const int bid = blockIdx.x;        // t*B_DIM + b
  const int t = bid >> 3;
  const int b = bid & 7;
  const float* mb = meta + b * 4;
  const float s0 = mb[0];
  const float s1 = mb[1];
  const int i0 = ((const int*)mb)[2];
  const int i1 = ((const int*)mb)[3];

  const size_t nstride = (size_t)T_DIM * B_DIM * D_DIM;
  const size_t rowOff  = ((size_t)t * B_DIM + b) * D_DIM + threadIdx.x * 4;

  float4 h0 = *(const float4*)(hs + (size_t)i0 * nstride + rowOff);
  float4 h1 = *(const float4*)(hs + (size_t)i1 * nstride + rowOff);
  float4 o;
  o.x = fmaf(s0, h0.x, s1 * h1.x);
  o.y = fmaf(s0, h0.y, s1 * h1.y);
  o.z = fmaf(s0, h0.z, s1 * h1.z);
  o.w = fmaf(s0, h0.w, s1 * h1.w);
  *(float4*)(out + rowOff) = o;
}

// ---------------------------------------------------------------------------
extern "C" void kernel_launch(void* const* d_in, const int* in_sizes, int n_in,
                              void* d_out, int out_size, void* d_ws, size_t ws_size,
                              hipStream_t stream) {
  (void)in_sizes; (void)n_in; (void)out_size; (void)ws_size;
  const float* hs  = (const float*)d_in[0];   // hidden_states
  const float* nmt = (const float*)d_in[1];   // nmt_state
  const float* gw  = (const float*)d_in[2];   // gate_w [1,512]
  const float* gb  = (const float*)d_in[3];   // gate_b [1]
  float* out = (float*)d_out;

  float* partial = (float*)d_ws;              // B_DIM*NCHUNKS*16 = 16384 floats (64 KB)
  float* meta    = partial + B_DIM * NCHUNKS * 16;  // 8*4 floats

  gate_partial_kernel<<<dim3(B_DIM * NCHUNKS), dim3(128), 0, stream>>>(hs, nmt, gw, partial);
  gate_finish_kernel<<<dim3(1), dim3(128), 0, stream>>>(partial, gb, meta);
  combine_kernel<<<dim3(T_DIM * B_DIM), dim3(256), 0, stream>>>(hs, meta, out);
}